// RoIAlign3D_59304908423360
// MI455X (gfx1250) — compile-verified
//
#include <hip/hip_runtime.h>
#include <hip/hip_bf16.h>

typedef __attribute__((ext_vector_type(2))) float v2f;
typedef __attribute__((ext_vector_type(8))) float v8f;

#define RN    256
#define CCH   128
#define DDIM  24
#define HDIM  128
#define WDIM  128
#define OUTHW 7
#define OUTD  4
#define WMAX  128   // max x-window (== WDIM)
#define PADC  17    // 16-wide fragment dim padded to 17 (coprime with 64 banks)

__device__ __forceinline__ int imin(int a, int b) { return a < b ? a : b; }
__device__ __forceinline__ int imax(int a, int b) { return a > b ? a : b; }

// One workgroup per (roi, od, oh). 8 wave32s; wave w handles channels [16w, 16w+16).
// Stage 1: G[x][c] = sum over 16 (z,y)-corner rows of w_zy * feat[row, x]  (coalesced)
// Stage 2: out[ow, c] = sum_x WxT[x][ow] * G[x][c]  via v_wmma_f32_16x16x4_f32
__global__ __launch_bounds__(256) void roialign3d_wmma_kernel(
    const float* __restrict__ feat,   // (2,128,24,128,128)
    const float* __restrict__ rois,   // (256,7)
    float* __restrict__ out)          // (256,128,4,7,7)
{
    __shared__ float sG[8][WMAX][PADC];  // per-wave interp rows, K-major: ~68 KB
    __shared__ float sWxT[WMAX][PADC];   // A-matrix, K-major (bins innermost): ~8.5 KB
    __shared__ int   sRow[16];           // (z,y) row element offsets
    __shared__ float sWzy[16];           // fused z*y weights (incl. validity + 0.25)
    __shared__ int   sMeta[3];           // xlo, wdp, batch

    const int r    = blockIdx.y;
    const int od   = blockIdx.x / OUTHW;
    const int oh   = blockIdx.x % OUTHW;
    const int tid  = threadIdx.x;
    const int wave = tid >> 5;
    const int lane = tid & 31;

    // zero WxT (scattered sparsely below)
    for (int i = tid; i < WMAX * PADC; i += 256) (&sWxT[0][0])[i] = 0.0f;
    __syncthreads();

    if (tid == 0) {
        const float* rp = rois + r * 7;
        const int   b  = (int)rp[0];
        const float x1 = rp[1] * 0.125f, y1 = rp[2] * 0.125f;
        const float x2 = rp[3] * 0.125f, y2 = rp[4] * 0.125f;
        const float z1 = rp[5] * 0.25f,  z2 = rp[6] * 0.25f;
        const float bw = fmaxf(x2 - x1, 1.0f) / (float)OUTHW;
        const float bh = fmaxf(y2 - y1, 1.0f) / (float)OUTHW;
        const float bd = fmaxf(z2 - z1, 1.0f) / (float)OUTD;

        int zi[4]; float zw[4];
        for (int s = 0; s < 2; ++s) {
            float c  = z1 + ((float)od + ((float)s + 0.5f) * 0.5f) * bd;
            bool  v  = (c > -1.0f) && (c < (float)DDIM);
            float cc = fminf(fmaxf(c, 0.0f), (float)(DDIM - 1));
            int   lo = imin((int)floorf(cc), DDIM - 1);
            int   hi = imin(lo + 1, DDIM - 1);
            float f  = cc - (float)lo;
            float vm = v ? 0.5f : 0.0f;               // 0.5 = SN-average factor for z
            zi[2*s] = lo; zw[2*s] = (1.0f - f) * vm;
            zi[2*s+1] = hi; zw[2*s+1] = f * vm;
        }
        int yi[4]; float yw[4];
        for (int s = 0; s < 2; ++s) {
            float c  = y1 + ((float)oh + ((float)s + 0.5f) * 0.5f) * bh;
            bool  v  = (c > -1.0f) && (c < (float)HDIM);
            float cc = fminf(fmaxf(c, 0.0f), (float)(HDIM - 1));
            int   lo = imin((int)floorf(cc), HDIM - 1);
            int   hi = imin(lo + 1, HDIM - 1);
            float f  = cc - (float)lo;
            float vm = v ? 0.5f : 0.0f;
            yi[2*s] = lo; yw[2*s] = (1.0f - f) * vm;
            yi[2*s+1] = hi; yw[2*s+1] = f * vm;
        }
        for (int a = 0; a < 4; ++a)
            for (int c2 = 0; c2 < 4; ++c2) {
                sRow[a*4 + c2] = (zi[a] * HDIM + yi[c2]) * WDIM;
                sWzy[a*4 + c2] = zw[a] * yw[c2];
            }

        // x samples: window extent, then scatter weights into WxT (K-major)
        int xmin = WDIM, xmax = -1;
        for (int ow = 0; ow < OUTHW; ++ow)
            for (int s = 0; s < 2; ++s) {
                float c = x1 + ((float)ow + ((float)s + 0.5f) * 0.5f) * bw;
                if (c > -1.0f && c < (float)WDIM) {
                    float cc = fminf(fmaxf(c, 0.0f), (float)(WDIM - 1));
                    int lo = imin((int)floorf(cc), WDIM - 1);
                    int hi = imin(lo + 1, WDIM - 1);
                    xmin = imin(xmin, lo); xmax = imax(xmax, hi);
                }
            }
        const int xlo = (xmax < 0) ? 0 : xmin;
        const int wdp = (xmax < 0) ? 4 : (((xmax - xlo + 1) + 3) & ~3);
        for (int ow = 0; ow < OUTHW; ++ow)
            for (int s = 0; s < 2; ++s) {
                float c = x1 + ((float)ow + ((float)s + 0.5f) * 0.5f) * bw;
                if (c > -1.0f && c < (float)WDIM) {
                    float cc = fminf(fmaxf(c, 0.0f), (float)(WDIM - 1));
                    int lo = imin((int)floorf(cc), WDIM - 1);
                    int hi = imin(lo + 1, WDIM - 1);
                    float f = cc - (float)lo;
                    sWxT[lo - xlo][ow] += 0.5f * (1.0f - f);
                    sWxT[hi - xlo][ow] += 0.5f * f;
                }
            }
        sMeta[0] = xlo; sMeta[1] = wdp; sMeta[2] = b;
    }
    __syncthreads();

    // wave-uniform params -> SGPRs (saddr-form loads, scalar FMA operands)
    const int xlo  = __builtin_amdgcn_readfirstlane(sMeta[0]);
    const int wdp  = __builtin_amdgcn_readfirstlane(sMeta[1]);
    const int bimg = __builtin_amdgcn_readfirstlane(sMeta[2]);
    const size_t DHW = (size_t)DDIM * HDIM * WDIM;
    const float* fb = feat + ((size_t)bimg * CCH + (size_t)wave * 16) * DHW;

    int rowo[16]; float rww[16];
    #pragma unroll
    for (int k = 0; k < 16; ++k) {
        rowo[k] = __builtin_amdgcn_readfirstlane(sRow[k]);
        rww[k]  = __int_as_float(__builtin_amdgcn_readfirstlane(__float_as_int(sWzy[k])));
    }

    // Stage 1: lanes stride along x -> coalesced 128B loads per row.
    // sG write addr = x*17 + c: gcd(17,64)=1 -> 32 lanes hit 32 distinct banks.
    for (int c = 0; c < 16; ++c) {
        const float* fc = fb + (size_t)c * DHW;
        for (int x = lane; x < wdp; x += 32) {
            const int xg = imin(xlo + x, WDIM - 1);   // pad region has zero Wx weight
            float acc = 0.0f;
            #pragma unroll
            for (int k = 0; k < 16; ++k)
                acc = fmaf(rww[k], fc[rowo[k] + xg], acc);
            sG[wave][x][c] = acc;
        }
    }
    __syncthreads();

    // Stage 2: D[bin, c] = A(WxT) x B(G), K chunked by 4.
    // Fragment read addr = k*17 + m: lanes 0-15 consecutive, lanes 16-31 offset
    // by 34 -> conflict-free.
    v8f dacc = {0.0f, 0.0f, 0.0f, 0.0f, 0.0f, 0.0f, 0.0f, 0.0f};
    const int m    = lane & 15;          // A: M row (bin); B/D: N col (channel)
    const int koff = (lane >> 4) << 1;   // lanes 16-31 hold K+2 (ISA 16x4 f32 layout)
    for (int kb = 0; kb < wdp; kb += 4) {
        v2f a, bf;
        a.x  = sWxT[kb + koff][m];
        a.y  = sWxT[kb + koff + 1][m];
        bf.x = sG[wave][kb + koff][m];
        bf.y = sG[wave][kb + koff + 1][m];
        dacc = __builtin_amdgcn_wmma_f32_16x16x4_f32(
            false, a, false, bf, (short)0, dacc, false, false);
    }

    // D VGPR j: lanes 0-15 -> (bin=j, c=lane); bins 7..15 are zero-weight padding
    if (lane < 16) {
        const int cg = wave * 16 + lane;
        const size_t ob = ((((size_t)r * CCH + cg) * OUTD + od) * OUTHW + oh) * OUTHW;
        #pragma unroll
        for (int j = 0; j < 7; ++j) out[ob + j] = dacc[j];
    }
}

extern "C" void kernel_launch(void* const* d_in, const int* in_sizes, int n_in,
                              void* d_out, int out_size, void* d_ws, size_t ws_size,
                              hipStream_t stream) {
    const float* feat = (const float*)d_in[0];
    const float* rois = (const float*)d_in[1];
    float* out = (float*)d_out;
    dim3 grid(OUTD * OUTHW, RN);   // 28 x 256 workgroups
    dim3 block(256);               // 8 wave32
    roialign3d_wmma_kernel<<<grid, block, 0, stream>>>(feat, rois, out);
}